// NeuroFuzzyANFIS_6674379178608
// MI455X (gfx1250) — compile-verified
//
#include <hip/hip_runtime.h>

// CDNA5 / gfx1250, wave32. Fused ANFIS forward:
//   one V_WMMA_F32_16X16X4_F32 chain (K=20, 5 steps) computes BOTH the
//   rule firing logits (cols 0..7) and y_r (cols 8..15) for 16 samples/wave.
typedef float v2f __attribute__((ext_vector_type(2)));
typedef float v8f __attribute__((ext_vector_type(8)));
typedef float v4f __attribute__((ext_vector_type(4)));

// ---- DPP8 xor-butterfly within groups of 8 lanes (pure VALU, no LDS) ----
constexpr unsigned dpp8_sel_xor(int x) {
    unsigned s = 0;
    for (int i = 0; i < 8; ++i) s |= (unsigned)((i ^ x) & 7) << (3 * i);
    return s;
}

template <unsigned SEL>
__device__ __forceinline__ float mov_dpp8_f(float v) {
    int r = __builtin_amdgcn_mov_dpp8(__builtin_bit_cast(int, v), (int)SEL);
    return __builtin_bit_cast(float, r);
}

__device__ __forceinline__ float reduce8_sum(float v) {
    v += mov_dpp8_f<dpp8_sel_xor(1)>(v);
    v += mov_dpp8_f<dpp8_sel_xor(2)>(v);
    v += mov_dpp8_f<dpp8_sel_xor(4)>(v);
    return v;
}

__global__ __launch_bounds__(256)
void anfis_wmma_kernel(const float* __restrict__ x,          // (B,8)
                       const float* __restrict__ c,          // (8,3)
                       const float* __restrict__ log_sigma,  // (8,3)
                       const float* __restrict__ cons,       // (8,9)
                       const int*   __restrict__ rules,      // (8,8)
                       float* __restrict__ y_out,            // (B)
                       float* __restrict__ w_out,            // (B,8)
                       float* __restrict__ yr_out,           // (B,8)
                       int B)
{
    const int lane  = threadIdx.x & 31;
    const int n     = lane & 15;          // D column: 0..7 = logit S, 8..15 = y_r rule (n-8)
    const int kt    = (lane >> 4) * 2;    // K sub-offset per half-wave (A/B 32-bit layout)
    const bool hi   = (lane & 16) != 0;
    const int gwave = blockIdx.x * (blockDim.x >> 5) + (threadIdx.x >> 5);
    const int nwave = gridDim.x * (blockDim.x >> 5);

    // ---------------- persistent B fragments (5 K-steps of 4) ----------------
    // B[k][n]:  n<8  : k==0 -> D0[n]; 1..8 -> Q[n][k-1]; 9..16 -> P[n][k-9]; else 0
    //           n>=8 : k<9  -> consequents[n-8][k]; else 0
    v2f bf[5];
#pragma unroll
    for (int s = 0; s < 5; ++s) {
        float bv[2];
#pragma unroll
        for (int j = 0; j < 2; ++j) {
            const int kk = s * 4 + kt + j;
            float v = 0.0f;
            if (n >= 8) {
                if (kk < 9) v = cons[(n - 8) * 9 + kk];
            } else {
                if (kk == 0) {                         // D0[r] = sum_m 0.5*c^2/sigma^2
                    float acc = 0.0f;
                    for (int m = 0; m < 8; ++m) {
                        const int   km  = rules[n * 8 + m];
                        const float sg  = __expf(log_sigma[m * 3 + km]) + 1e-6f;
                        const float is2 = 1.0f / (sg * sg);
                        const float cv  = c[m * 3 + km];
                        acc += 0.5f * is2 * cv * cv;
                    }
                    v = acc;
                } else if (kk <= 8) {                  // Q[r][m] = -c/sigma^2
                    const int   m   = kk - 1;
                    const int   km  = rules[n * 8 + m];
                    const float sg  = __expf(log_sigma[m * 3 + km]) + 1e-6f;
                    v = -(1.0f / (sg * sg)) * c[m * 3 + km];
                } else if (kk <= 16) {                 // P[r][m] = 0.5/sigma^2
                    const int   m   = kk - 9;
                    const int   km  = rules[n * 8 + m];
                    const float sg  = __expf(log_sigma[m * 3 + km]) + 1e-6f;
                    v = 0.5f / (sg * sg);
                }
            }
            bv[j] = v;
        }
        bf[s].x = bv[0];
        bf[s].y = bv[1];
    }

    const int ntiles = B >> 4;            // 16 samples per wave-tile

    for (int tile = gwave; tile < ntiles; tile += nwave) {
        const long base = (long)tile * 16;

        // Each lane loads its A row as two b128s (32B/row, branch-free).
        const v4f* xv = (const v4f*)(x + (base + n) * 8);
        const v4f lo4 = xv[0];
        const v4f hi4 = xv[1];

        if (tile + nwave < ntiles)
            __builtin_prefetch(x + (base + (long)nwave * 16 + n) * 8, 0, 1);

        const float x0 = lo4.x, x1 = lo4.y, x2 = lo4.z, x3 = lo4.w;
        const float x4 = hi4.x, x5 = hi4.y, x6 = hi4.z, x7 = hi4.w;

        // A row (K=20): [1, x0..x7, x0^2..x7^2, 0,0,0]
        // kt==0 lanes supply K = {0,1},{4,5},{8,9},{12,13},{16,17}
        // kt==2 lanes supply K = {2,3},{6,7},{10,11},{14,15},{18,19}
        v2f af[5];
        af[0].x = hi ? x1 : 1.0f;         af[0].y = hi ? x2 : x0;
        af[1].x = hi ? x5 : x3;           af[1].y = hi ? x6 : x4;
        af[2].x = hi ? x1 * x1 : x7;      af[2].y = hi ? x2 * x2 : x0 * x0;
        af[3].x = hi ? x5 * x5 : x3 * x3; af[3].y = hi ? x6 * x6 : x4 * x4;
        af[4].x = hi ? 0.0f : x7 * x7;    af[4].y = 0.0f;

        // D = A(16x20) x B(20x16): cols 0..7 = S logits, cols 8..15 = y_r
        v8f acc = {0.f, 0.f, 0.f, 0.f, 0.f, 0.f, 0.f, 0.f};
#pragma unroll
        for (int s = 0; s < 5; ++s)
            acc = __builtin_amdgcn_wmma_f32_16x16x4_f32(
                false, af[s], false, bf[s], (short)0, acc, false, false);

        // ---- epilogue pass 1: full-EXEC compute (DPP/bpermute need all lanes) ----
        float wns[8];   // per D-reg: w_norm           (valid lanes n<8)
        float yss[8];   // per D-reg: reduced y        (valid lanes n>=8, replicated)
#pragma unroll
        for (int v = 0; v < 8; ++v) {
            const float val = acc[v];
            const float fir = __expf(-val);                 // junk for n>=8, contained
            const float tot = reduce8_sum(fir);             // DPP8 butterfly in 8-lane groups
            const float wn  = fir * __builtin_amdgcn_rcpf(tot + 1e-8f);
            // single cross-group exchange: lanes 8..15 receive their rule's w_norm
            const float wnp = __shfl_xor(wn, 8, 32);
            const float ys  = reduce8_sum(wnp * val);       // valid within lanes 8..15 group
            wns[v] = wn;
            yss[v] = ys;
        }

        // ---- epilogue pass 2: single divergence region for all stores ----
        const long bb = base + ((lane >> 4) << 3);          // first sample of this half-wave
        if (n < 8) {
            // lanes 0..7 (16..23): w_norm, 8 consecutive floats per group per v
#pragma unroll
            for (int v = 0; v < 8; ++v)
                __builtin_nontemporal_store(wns[v], &w_out[(bb + v) * 8 + n]);
        } else {
            // lanes 8..15 (24..31): y_r straight out of the accumulator
#pragma unroll
            for (int v = 0; v < 8; ++v)
                __builtin_nontemporal_store(acc[v], &yr_out[(bb + v) * 8 + (n - 8)]);
            if (n == 8) {
                // lane 8 (24): 8 consecutive y values -> two b128 stores
                v4f ylo, yhi;
                ylo.x = yss[0]; ylo.y = yss[1]; ylo.z = yss[2]; ylo.w = yss[3];
                yhi.x = yss[4]; yhi.y = yss[5]; yhi.z = yss[6]; yhi.w = yss[7];
                __builtin_nontemporal_store(ylo, (v4f*)(y_out + bb));
                __builtin_nontemporal_store(yhi, (v4f*)(y_out + bb + 4));
            }
        }
    }
}

extern "C" void kernel_launch(void* const* d_in, const int* in_sizes, int n_in,
                              void* d_out, int out_size, void* d_ws, size_t ws_size,
                              hipStream_t stream) {
    const float* x     = (const float*)d_in[0];
    const float* c     = (const float*)d_in[1];
    const float* lsig  = (const float*)d_in[2];
    const float* cons  = (const float*)d_in[3];
    const int*   rules = (const int*)d_in[4];

    const int B = in_sizes[0] / 8;               // 1048576

    float* y  = (float*)d_out;                   // (B)
    float* w  = y + (size_t)B;                   // (B,8)
    float* yr = y + (size_t)B * 9;               // (B,8)

    const int ntiles = B >> 4;                   // 65536 wave-tiles
    int blocks = (ntiles + 7) / 8;               // 8 waves/block
    if (blocks > 1024) blocks = 1024;            // grid-stride: ~8 tiles/wave amortizes setup
    if (blocks < 1) blocks = 1;

    anfis_wmma_kernel<<<blocks, 256, 0, stream>>>(x, c, lsig, cons, rules, y, w, yr, B);
}